// GCN_79860621902168
// MI455X (gfx1250) — compile-verified
//
#include <hip/hip_runtime.h>

typedef __attribute__((ext_vector_type(2))) float v2f;
typedef __attribute__((ext_vector_type(8))) float v8f;

#define HDIM 128
#define LDA  132   // padded LDS stride: 132 % 64 = 4 -> rows land on distinct banks

// ---------- trivial helpers ----------
__global__ __launch_bounds__(256) void k_zero(float* p, int n) {
    int i = blockIdx.x * 256 + threadIdx.x;
    if (i < n) p[i] = 0.0f;
}

__global__ __launch_bounds__(256) void k_deg(const int* __restrict__ dst, float* deg, int E) {
    int e = blockIdx.x * 256 + threadIdx.x;
    if (e < E) atomicAdd(&deg[dst[e]], 1.0f);
}

__global__ __launch_bounds__(256) void k_dinv(float* deg, int n) {
    int i = blockIdx.x * 256 + threadIdx.x;
    if (i < n) deg[i] = rsqrtf(deg[i] + 1.0f);   // deg includes self-loop (+1)
}

// Wt[c][k] = W[k][c]   (128x128)
__global__ __launch_bounds__(256) void k_transpose(const float* __restrict__ W, float* __restrict__ Wt) {
    int i = blockIdx.x * 256 + threadIdx.x;   // 0..16383
    int r = i >> 7, c = i & 127;
    Wt[c * HDIM + r] = W[r * HDIM + c];
}

// ---------- WMMA fp32 GEMM: O[16 rows x 128 cols] per block ----------
// A: [N,128] row-major, Wt: [128,128] (column of W per row), O: [N,128]
__global__ __launch_bounds__(256) void k_gemm_wmma(const float* __restrict__ A,
                                                   const float* __restrict__ Wt,
                                                   float* __restrict__ O) {
    __shared__ float lA[16 * LDA];
    const int rowBase = blockIdx.x * 16;

    // Stage the 16x128 A strip into padded LDS via the gfx1250 async-to-LDS pipe.
    // Each thread moves 8 contiguous floats (2 x b128). 16B alignment holds on
    // both sides: global offset = 32B multiple, LDS offset = r*528 + c0*4.
    {
        int t  = threadIdx.x;
        int r  = t >> 4;           // 0..15
        int c0 = (t & 15) * 8;     // 0..120
        const float* gsrc = A + (size_t)(rowBase + r) * HDIM + c0;
        unsigned     loff = (unsigned)(uintptr_t)(&lA[r * LDA + c0]);
        asm volatile("global_load_async_to_lds_b128 %0, %1, off\n\t"
                     "global_load_async_to_lds_b128 %2, %3, off"
                     :
                     : "v"(loff), "v"(gsrc), "v"(loff + 16u), "v"(gsrc + 4)
                     : "memory");
    }
    asm volatile("s_wait_asynccnt 0x0" ::: "memory");   // this wave's async copies done
    __syncthreads();                                    // publish tile to all 8 waves

    const int wave    = threadIdx.x >> 5;   // 8 waves -> 8 col tiles of 16
    const int lane    = threadIdx.x & 31;
    const int hi      = lane >> 4;          // 0: K pair {0,1}; 1: K pair {2,3}
    const int lo      = lane & 15;          // M index (A) / N index (B,D)
    const int colBase = wave * 16;

    const float* wtCol = Wt + (size_t)(colBase + lo) * HDIM;

    v8f acc = {};
#pragma unroll
    for (int kk = 0; kk < 32; ++kk) {       // K = 128 = 32 * 4
        int kb = kk * 4 + hi * 2;
        v2f a, b;
        a.x = lA[lo * LDA + kb];
        a.y = lA[lo * LDA + kb + 1];
        b.x = wtCol[kb];
        b.y = wtCol[kb + 1];
        acc = __builtin_amdgcn_wmma_f32_16x16x4_f32(false, a, false, b,
                                                    (short)0, acc, false, false);
    }

    // D layout: VGPR r -> M = r + 8*hi, N = lo
    float* out = O + (size_t)(rowBase + hi * 8) * HDIM + colBase + lo;
#pragma unroll
    for (int r = 0; r < 8; ++r)
        out[(size_t)r * HDIM] = acc[r];
}

// agg[n][f] = h[n][f] * dinv[n]^2   (self-loop term, also initializes accumulator)
__global__ __launch_bounds__(256) void k_selfloop(const float* __restrict__ h,
                                                  const float* __restrict__ dinv,
                                                  float* __restrict__ agg, int total) {
    int i = blockIdx.x * 256 + threadIdx.x;
    if (i < total) {
        float di = dinv[i >> 7];
        agg[i] = h[i] * di * di;
    }
}

// one wave per edge; lane handles 4 features (float4)
__global__ __launch_bounds__(256) void k_edge_agg(const int* __restrict__ src,
                                                  const int* __restrict__ dst,
                                                  const float* __restrict__ dinv,
                                                  const float* __restrict__ h,
                                                  float* __restrict__ agg, int E) {
    int e = blockIdx.x * 8 + (threadIdx.x >> 5);
    if (e >= E) return;
    int lane = threadIdx.x & 31;
    int s = src[e], d = dst[e];
    float nrm = dinv[s] * dinv[d];
    float4 v = ((const float4*)(h + (size_t)s * HDIM))[lane];
    float* a = agg + (size_t)d * HDIM + lane * 4;
    atomicAdd(a + 0, v.x * nrm);
    atomicAdd(a + 1, v.y * nrm);
    atomicAdd(a + 2, v.z * nrm);
    atomicAdd(a + 3, v.w * nrm);
}

__global__ __launch_bounds__(256) void k_bias_act(float* agg, const float* __restrict__ b,
                                                  int total, int relu) {
    int i = blockIdx.x * 256 + threadIdx.x;
    if (i < total) {
        float v = agg[i] + b[i & 127];
        if (relu) v = fmaxf(v, 0.0f);
        agg[i] = v;
    }
}

__global__ __launch_bounds__(256) void k_pool(const float* __restrict__ h,
                                              const int* __restrict__ batch,
                                              float* gsum, float* gcnt, int total) {
    int i = blockIdx.x * 256 + threadIdx.x;
    if (i < total) {
        int n = i >> 7, f = i & 127;
        int g = batch[n];
        atomicAdd(&gsum[g * HDIM + f], h[i]);
        if (f == 0) atomicAdd(&gcnt[g], 1.0f);
    }
}

__global__ __launch_bounds__(64) void k_classify(const float* __restrict__ gsum,
                                                 const float* __restrict__ gcnt,
                                                 const float* __restrict__ Wc,
                                                 const float* __restrict__ bc,
                                                 float* __restrict__ out, int G, int C) {
    int i = blockIdx.x * 64 + threadIdx.x;
    if (i >= G * C) return;
    int g = i / C, c = i % C;
    float inv = 1.0f / fmaxf(gcnt[g], 1.0f);
    float acc = bc[c];
#pragma unroll 8
    for (int k = 0; k < HDIM; ++k)
        acc += gsum[g * HDIM + k] * inv * Wc[k * C + c];
    out[i] = acc;
}

extern "C" void kernel_launch(void* const* d_in, const int* in_sizes, int n_in,
                              void* d_out, int out_size, void* d_ws, size_t ws_size,
                              hipStream_t stream) {
    const float* x    = (const float*)d_in[0];
    const int*   eidx = (const int*)  d_in[1];
    const int*   batch= (const int*)  d_in[2];
    const float* W1   = (const float*)d_in[3];
    const float* b1   = (const float*)d_in[4];
    const float* W2   = (const float*)d_in[5];
    const float* b2   = (const float*)d_in[6];
    const float* W3   = (const float*)d_in[7];
    const float* b3   = (const float*)d_in[8];
    const float* Wc   = (const float*)d_in[9];
    const float* bc   = (const float*)d_in[10];
    float* out = (float*)d_out;

    const int N = in_sizes[0] / HDIM;
    const int E = in_sizes[1] / 2;
    const int C = in_sizes[10];
    const int G = out_size / C;
    const int* src = eidx;
    const int* dst = eidx + E;

    float* ws = (float*)d_ws;
    const size_t NH = (size_t)N * HDIM;
    float* bufA = ws;                 // h = X @ W
    float* bufB = bufA + NH;          // aggregated output (also next layer input)
    float* dinv = bufB + NH;          // degree, then rsqrt(deg+1)
    float* wt   = dinv + N;           // transposed weight (128x128)
    float* gsum = wt + HDIM * HDIM;   // G x 128
    float* gcnt = gsum + (size_t)G * HDIM;

    dim3 b256(256);
    const int total = N * HDIM;

    // degrees -> dinv (in place)
    k_zero<<<(N + 255) / 256, b256, 0, stream>>>(dinv, N);
    k_deg <<<(E + 255) / 256, b256, 0, stream>>>(dst, dinv, E);
    k_dinv<<<(N + 255) / 256, b256, 0, stream>>>(dinv, N);

    const float* Ws[3] = {W1, W2, W3};
    const float* bs[3] = {b1, b2, b3};
    const int    rl[3] = {1, 1, 0};
    const float* inbuf = x;
    for (int l = 0; l < 3; ++l) {
        k_transpose<<<64, b256, 0, stream>>>(Ws[l], wt);
        k_gemm_wmma<<<N / 16, b256, 0, stream>>>(inbuf, wt, bufA);
        k_selfloop <<<(total + 255) / 256, b256, 0, stream>>>(bufA, dinv, bufB, total);
        k_edge_agg <<<(E + 7) / 8, b256, 0, stream>>>(src, dst, dinv, bufA, bufB, E);
        k_bias_act <<<(total + 255) / 256, b256, 0, stream>>>(bufB, bs[l], total, rl[l]);
        inbuf = bufB;
    }

    // global mean pool + classifier
    k_zero<<<(G * HDIM + G + 255) / 256, b256, 0, stream>>>(gsum, G * HDIM + G); // gcnt contiguous
    k_pool<<<(total + 255) / 256, b256, 0, stream>>>(bufB, batch, gsum, gcnt, total);
    k_classify<<<(G * C + 63) / 64, 64, 0, stream>>>(gsum, gcnt, Wc, bc, out, G, C);
}